// GNN_Overlapping_44220983280305
// MI455X (gfx1250) — compile-verified
//
#include <hip/hip_runtime.h>
#include <hip/hip_bf16.h>
#include <math.h>

// ---------------------------------------------------------------------------
// GCN: 2x (fp32 WMMA GEMM -> edge gather/scale/scatter-add -> bias+ReLU),
//      then fp32 WMMA FC with fused bias + sigmoid.
// Memory-bound by edge aggregation (~1.8 GB HBM traffic); GEMMs kept in
// full-precision v_wmma_f32_16x16x4_f32 since their FLOPs are negligible.
// ---------------------------------------------------------------------------

typedef float v2f __attribute__((ext_vector_type(2)));
typedef float v8f __attribute__((ext_vector_type(8)));

#define HIDDIM 64

// ---- degree / normalization ------------------------------------------------

__global__ void deg_init_kernel(float* __restrict__ deg, int n) {
    int i = blockIdx.x * blockDim.x + threadIdx.x;
    if (i < n) deg[i] = 1.0f;  // self-loop contribution
}

__global__ void deg_accum_kernel(const int* __restrict__ edge_index,
                                 float* __restrict__ deg, int E) {
    int e = blockIdx.x * blockDim.x + threadIdx.x;
    if (e < E) {
        int dst = edge_index[E + e];  // col = target
        atomicAdd(&deg[dst], 1.0f);
    }
}

__global__ void dinv_kernel(float* __restrict__ deg, int n) {
    int i = blockIdx.x * blockDim.x + threadIdx.x;
    if (i < n) {
        float d = deg[i];            // >= 1 always (self loops)
        deg[i] = __frsqrt_rn(d);     // in-place: deg becomes dinv
    }
}

// ---- fp32 WMMA GEMM: C[M x OUTC] = A[M x K] @ W[K x OUTC] ------------------
// One wave computes one 16x16 output tile, accumulating K in steps of 4 with
// v_wmma_f32_16x16x4_f32.  ACT: 0 = none, 2 = bias + sigmoid (FC head).

template <int K, int OUTC, int ACT>
__global__ void wmma_gemm_kernel(const float* __restrict__ A,
                                 const float* __restrict__ W,
                                 const float* __restrict__ bias,
                                 float* __restrict__ C, int M) {
    const int lane   = threadIdx.x & 31;
    const int waveId = threadIdx.x >> 5;
    const int wavesPerBlk = blockDim.x >> 5;
    const int tilesN = OUTC / 16;
    const int tilesM = M / 16;          // M is a multiple of 16 (100000)
    const int nTiles = tilesM * tilesN;

    const int laneRow = lane & 15;      // row/col within tile for this lane
    const int kHalf   = (lane >> 4) * 2; // 0 for lanes 0-15, 2 for lanes 16-31

    for (int t = blockIdx.x * wavesPerBlk + waveId; t < nTiles;
         t += gridDim.x * wavesPerBlk) {
        const int tm = t / tilesN;
        const int tn = t - tm * tilesN;

        const int row     = tm * 16 + laneRow;
        const int colBase = tn * 16 + laneRow;
        const float* __restrict__ arow = A + (size_t)row * K;

        v8f acc = {};
#pragma unroll
        for (int k0 = 0; k0 < K; k0 += 4) {
            // A fragment (16x4): lanes 0-15 -> K = k0+0/1, lanes 16-31 -> k0+2/3
            v2f a;
            a.x = arow[k0 + kHalf + 0];
            a.y = arow[k0 + kHalf + 1];
            // B fragment (4x16): VGPR0 = rows k0+0 / k0+2, VGPR1 = k0+1 / k0+3
            v2f b;
            b.x = W[(size_t)(k0 + kHalf + 0) * OUTC + colBase];
            b.y = W[(size_t)(k0 + kHalf + 1) * OUTC + colBase];
            acc = __builtin_amdgcn_wmma_f32_16x16x4_f32(
                false, a, false, b, (short)0, acc, false, false);
        }

        // C/D layout: VGPR r -> M = r (lanes 0-15) / r+8 (lanes 16-31), N = lane&15
        const int rbase = tm * 16 + (lane >> 4) * 8;
#pragma unroll
        for (int r = 0; r < 8; ++r) {
            float v = acc[r];
            if (ACT == 2) {
                v += bias[colBase];
                v = 1.0f / (1.0f + __expf(-v));   // sigmoid
            }
            C[(size_t)(rbase + r) * OUTC + colBase] = v;
        }
    }
}

// ---- aggregation -----------------------------------------------------------
// Init: AGG[i,:] = T[i,:] * dinv[i]^2   (self-loop term; also zero-inits ws)
__global__ void agg_init_kernel(const float* __restrict__ T,
                                const float* __restrict__ dinv,
                                float* __restrict__ AGG, int n) {
    int gid = blockIdx.x * blockDim.x + threadIdx.x;   // n*32 threads, float2 each
    if (gid >= n * 32) return;
    const int node = gid >> 5;
    const int f2   = gid & 31;
    const float d  = dinv[node];
    const float w  = d * d;
    const float2* __restrict__ src = (const float2*)(T + (size_t)node * HIDDIM) + f2;
    float2 v = *src;
    float2* __restrict__ out = (float2*)(AGG + (size_t)node * HIDDIM) + f2;
    *out = make_float2(v.x * w, v.y * w);
}

// Scatter: one wave per edge; 32 lanes x float2 = one 256B feature row.
// AGG[dst,:] += T[src,:] * dinv[src] * dinv[dst]   (atomic f32 adds)
__global__ void edge_scatter_kernel(const int* __restrict__ edge_index,
                                    const float* __restrict__ T,
                                    const float* __restrict__ dinv,
                                    float* __restrict__ AGG, int E) {
    int gid = blockIdx.x * blockDim.x + threadIdx.x;
    int e   = gid >> 5;
    if (e >= E) return;
    const int lane = gid & 31;
    const int src  = edge_index[e];
    const int dst  = edge_index[E + e];
    const float w  = dinv[src] * dinv[dst];
    const float2* __restrict__ t = (const float2*)(T + (size_t)src * HIDDIM) + lane;
    float2 v = *t;
    float* __restrict__ o = AGG + (size_t)dst * HIDDIM + lane * 2;
    atomicAdd(o + 0, v.x * w);
    atomicAdd(o + 1, v.y * w);
}

// H = relu(AGG + b), in place
__global__ void bias_relu_kernel(float* __restrict__ H,
                                 const float* __restrict__ b, int n) {
    int gid = blockIdx.x * blockDim.x + threadIdx.x;   // n*64 threads
    if (gid >= n * HIDDIM) return;
    const int c = gid & (HIDDIM - 1);
    float v = H[gid] + b[c];
    H[gid] = v > 0.0f ? v : 0.0f;
}

// ---------------------------------------------------------------------------

extern "C" void kernel_launch(void* const* d_in, const int* in_sizes, int n_in,
                              void* d_out, int out_size, void* d_ws, size_t ws_size,
                              hipStream_t stream) {
    const float* x          = (const float*)d_in[0];
    const int*   edge_index = (const int*)d_in[1];
    const float* W1         = (const float*)d_in[2];
    const float* b1         = (const float*)d_in[3];
    const float* W2         = (const float*)d_in[4];
    const float* b2         = (const float*)d_in[5];
    const float* Wfc        = (const float*)d_in[6];
    const float* bfc        = (const float*)d_in[7];
    float*       out        = (float*)d_out;

    const int N = in_sizes[0] / 128;   // 100000
    const int E = in_sizes[1] / 2;     // 1600000

    // workspace: dinv[N] | bufA[N*64] | bufB[N*64]
    float* dinv = (float*)d_ws;
    float* bufA = dinv + N;
    float* bufB = bufA + (size_t)N * HIDDIM;

    const int TPB = 256;

    // ---- degree / dinv ----
    deg_init_kernel<<<(N + TPB - 1) / TPB, TPB, 0, stream>>>(dinv, N);
    deg_accum_kernel<<<(E + TPB - 1) / TPB, TPB, 0, stream>>>(edge_index, dinv, E);
    dinv_kernel<<<(N + TPB - 1) / TPB, TPB, 0, stream>>>(dinv, N);

    const int wavesPerBlk = TPB / 32;
    const int scatterBlks = (E * 32 + TPB - 1) / TPB;
    const int initBlks    = (N * 32 + TPB - 1) / TPB;
    const int reluBlks    = (N * HIDDIM + TPB - 1) / TPB;

    // ---- layer 1: T = X @ W1 ----
    {
        const int tiles = (N / 16) * (HIDDIM / 16);
        const int blks  = (tiles + wavesPerBlk - 1) / wavesPerBlk;
        wmma_gemm_kernel<128, HIDDIM, 0><<<blks, TPB, 0, stream>>>(x, W1, nullptr, bufA, N);
    }
    agg_init_kernel<<<initBlks, TPB, 0, stream>>>(bufA, dinv, bufB, N);
    edge_scatter_kernel<<<scatterBlks, TPB, 0, stream>>>(edge_index, bufA, dinv, bufB, E);
    bias_relu_kernel<<<reluBlks, TPB, 0, stream>>>(bufB, b1, N);

    // ---- layer 2: T = H1 @ W2 ----
    {
        const int tiles = (N / 16) * (HIDDIM / 16);
        const int blks  = (tiles + wavesPerBlk - 1) / wavesPerBlk;
        wmma_gemm_kernel<HIDDIM, HIDDIM, 0><<<blks, TPB, 0, stream>>>(bufB, W2, nullptr, bufA, N);
    }
    agg_init_kernel<<<initBlks, TPB, 0, stream>>>(bufA, dinv, bufB, N);
    edge_scatter_kernel<<<scatterBlks, TPB, 0, stream>>>(edge_index, bufA, dinv, bufB, E);
    bias_relu_kernel<<<reluBlks, TPB, 0, stream>>>(bufB, b2, N);

    // ---- FC head: out = sigmoid(H2 @ Wfc + bfc), 32 cols ----
    {
        const int tiles = (N / 16) * (32 / 16);
        const int blks  = (tiles + wavesPerBlk - 1) / wavesPerBlk;
        wmma_gemm_kernel<HIDDIM, 32, 2><<<blks, TPB, 0, stream>>>(bufB, Wfc, bfc, out, N);
    }
}